// JointMembership_FuzzyLayer_JAFFE_NoPool_89816356094189
// MI455X (gfx1250) — compile-verified
//
#include <hip/hip_runtime.h>
#include <hip/hip_bf16.h>

// Problem constants (from reference): B=2048, P=512, C=7, HW=32
//   NBP = B*P = 1,048,576 pixel-pair rows; out = [B, 2P, C] -> 14 contiguous f32 per bp-row
// Simplified math:
//   z0_c = cos(ry_c)*cos(x0) - sin(ry_c)*cos(rz_c)*sin(x0)*sin(x1)
//   z1   = cos(x0)*cos(x1)            (class independent)
//   out  = (z+1)/2
// => out_row[0:14] = F[0:4] x W[4:14] with
//   F = [cos x0, sin x0*sin x1, cos x0*cos x1, 1]
//   W[:,n<7]     = [0.5*cos(ry_n), -0.5*sin(ry_n)*cos(rz_n), 0,   0.5]
//   W[:,7<=n<14] = [0,             0,                        0.5, 0.5]
// One V_WMMA_F32_16X16X4_F32 per 16 bp rows; 8 tiles per wave, B matrix hoisted.

typedef __attribute__((ext_vector_type(2))) float v2f;
typedef __attribute__((ext_vector_type(8))) float v8f;

#define NPIX 1024            // 32*32 pixels per batch image
#define P_LOG2 9             // P = 512
#define NBP (2048 * 512)     // total bp rows
#define WAVES_PER_BLOCK 8    // 256 threads, wave32
#define TILES_PER_WAVE 8     // 128 bp rows / wave, 7168 B contiguous store region

__global__ __launch_bounds__(256) void qcirc_wmma_kernel(
    const float* __restrict__ x,        // [B, 1, 32, 32]
    const float* __restrict__ rz,       // [7]
    const float* __restrict__ ry,       // [7]
    const int*   __restrict__ pidx,     // [B, P, 2]
    float*       __restrict__ out)      // [B, 2P, 7]
{
    const int lane  = threadIdx.x & 31;
    const int wave  = threadIdx.x >> 5;
    const int m     = lane & 15;        // row-in-tile / column index
    const bool hi   = lane >= 16;       // high half-wave holds K={2,3}
    const int tile0 = (blockIdx.x * WAVES_PER_BLOCK + wave) * TILES_PER_WAVE;
    const int bp0   = tile0 << 4;       // first bp row of this wave

    // ---- B matrix, 4x16 f32 (K=row, N=column = lane&15); tile-invariant, built once
    const int   n   = m;
    const float ryv = (n < 7) ? ry[n] : 0.0f;
    const float rzv = (n < 7) ? rz[n] : 0.0f;
    const float w0  = (n < 7) ?  0.5f * __cosf(ryv)               : 0.0f;
    const float w1  = (n < 7) ? -0.5f * __sinf(ryv) * __cosf(rzv) : 0.0f;
    const float w2  = (n >= 7 && n < 14) ? 0.5f : 0.0f;
    const float w3  = (n < 14)           ? 0.5f : 0.0f;
    v2f bm;
    bm.x = hi ? w2 : w0;                // K=0 (lo) / K=2 (hi)
    bm.y = hi ? w3 : w1;                // K=1 (lo) / K=3 (hi)

    // ---- convergent compute phase: 8 tiles, EXEC all-1s around every WMMA
    v8f acc[TILES_PER_WAVE];
    #pragma unroll
    for (int t = 0; t < TILES_PER_WAVE; ++t) {
        const int bp = bp0 + (t << 4) + m;
        const int b  = bp >> P_LOG2;                // bp == b*P + p
        const int i0 = pidx[2 * bp + 0];
        const int i1 = pidx[2 * bp + 1];
        const float* xb = x + b * NPIX;
        const float x0 = xb[i0];
        const float x1 = xb[i1];
        // HW transcendentals: v_sin_f32 / v_cos_f32 (inputs ~N(0,1), in range)
        const float s0 = __sinf(x0), c0 = __cosf(x0);
        const float s1 = __sinf(x1), c1 = __cosf(x1);

        // A matrix, 16x4 f32: lanes 0-15 -> K={0,1}, lanes 16-31 -> K={2,3}
        v2f a;
        a.x = hi ? (c0 * c1) : c0;          // K=0: cos x0      | K=2: cos x0 * cos x1
        a.y = hi ? 1.0f      : (s0 * s1);   // K=1: sinx0*sinx1 | K=3: 1

        v8f c = {};
        acc[t] = __builtin_amdgcn_wmma_f32_16x16x4_f32(
            /*neg_a=*/false, a, /*neg_b=*/false, bm,
            /*c_mod=*/(short)0, c, /*reuse_a=*/false, /*reuse_b=*/false);
    }

    // ---- store phase: D VGPR v = (M = v + hi*8, N = n); out row stride = 14 floats
    // Divergent region contains no WMMA; 64 dword stores over a contiguous 7 KiB span.
    if (n < 14) {
        const int rowBase0 = (bp0 + (hi ? 8 : 0)) * 14 + n;
        #pragma unroll
        for (int t = 0; t < TILES_PER_WAVE; ++t) {
            const int rowBase = rowBase0 + t * (16 * 14);
            #pragma unroll
            for (int v = 0; v < 8; ++v)
                out[rowBase + v * 14] = acc[t][v];
        }
    }
}

extern "C" void kernel_launch(void* const* d_in, const int* in_sizes, int n_in,
                              void* d_out, int out_size, void* d_ws, size_t ws_size,
                              hipStream_t stream) {
    // setup_inputs() order: x, rz_params, ry_params, pair_idx
    const float* x    = (const float*)d_in[0];
    const float* rz   = (const float*)d_in[1];
    const float* ry   = (const float*)d_in[2];
    const int*   pidx = (const int*)d_in[3];
    float*       out  = (float*)d_out;

    const int tiles  = NBP / 16;                                   // 65536
    const int blocks = tiles / (WAVES_PER_BLOCK * TILES_PER_WAVE); // 1024
    qcirc_wmma_kernel<<<blocks, 256, 0, stream>>>(x, rz, ry, pidx, out);
}